// Decagon_31258771980722
// MI455X (gfx1250) — compile-verified
//
#include <hip/hip_runtime.h>

#define N_NODES 50000
#define N_EDGES 800000
#define DIM 128
#define N_REL 86
#define BATCH 2048

typedef float v2f __attribute__((ext_vector_type(2)));
typedef float v8f __attribute__((ext_vector_type(8)));

// ---------------------------------------------------------------- CSR build
__global__ void k_zero_cnt(int* __restrict__ cnt) {
  int i = blockIdx.x * blockDim.x + threadIdx.x;
  if (i < N_NODES) cnt[i] = 0;
}

__global__ void k_count(const int* __restrict__ dst, int* __restrict__ cnt) {
  int e = blockIdx.x * blockDim.x + threadIdx.x;
  if (e < N_EDGES) atomicAdd(&cnt[dst[e]], 1);
}

// single-block exclusive scan of cnt -> offs (offs[N_NODES] = total)
__global__ void k_scan(const int* __restrict__ cnt, int* __restrict__ offs) {
  __shared__ int sums[1024];
  const int tid = threadIdx.x;
  const int chunk = (N_NODES + 1023) / 1024;
  const int base = tid * chunk;
  int local = 0;
  for (int i = 0; i < chunk; ++i) {
    int idx = base + i;
    if (idx < N_NODES) local += cnt[idx];
  }
  sums[tid] = local;
  __syncthreads();
  for (int off = 1; off < 1024; off <<= 1) {
    int v = (tid >= off) ? sums[tid - off] : 0;
    __syncthreads();
    sums[tid] += v;
    __syncthreads();
  }
  int prefix = (tid == 0) ? 0 : sums[tid - 1];
  for (int i = 0; i < chunk; ++i) {
    int idx = base + i;
    if (idx < N_NODES) { offs[idx] = prefix; prefix += cnt[idx]; }
  }
  if (tid == 1023) offs[N_NODES] = prefix;
}

__global__ void k_dinv_cursor(const int* __restrict__ cnt, const int* __restrict__ offs,
                              float* __restrict__ dinv, int* __restrict__ cursor) {
  int i = blockIdx.x * blockDim.x + threadIdx.x;
  if (i < N_NODES) {
    dinv[i] = rsqrtf((float)(cnt[i] + 1));  // deg includes self-loop, always >= 1
    cursor[i] = offs[i];
  }
}

__global__ void k_scatter(const int* __restrict__ src, const int* __restrict__ dst,
                          int* __restrict__ cursor, int* __restrict__ csr) {
  int e = blockIdx.x * blockDim.x + threadIdx.x;
  if (e < N_EDGES) {
    int pos = atomicAdd(&cursor[dst[e]], 1);
    csr[pos] = src[e];
  }
}

// ---------------------------------------------------------------- fp32 WMMA GEMM
// C[M x 128] = A[M x 128] @ B[128 x 128] (optionally B^T), batched.
// One wave computes a 16x128 stripe via V_WMMA_F32_16X16X4_F32.
// A 16x4 layout: lanes 0-15 row=lane K={0,1}; lanes 16-31 row=lane-16 K={2,3}.
// B 4x16 layout mirrors it: lanes 0-15 col=lane K={0,1}; lanes 16-31 K={2,3}.
// C 16x16:  vgpr i, lanes 0-15 -> row i, lanes 16-31 -> row 8+i; col = lane&15.
__global__ __launch_bounds__(256) void k_wmma_gemm(
    const float* __restrict__ A, unsigned long long aStride,
    const float* __restrict__ B, unsigned long long bStride, int transB,
    float* __restrict__ C, unsigned long long cStride, int M) {
  __shared__ float Bs[DIM * 130];  // [col][k], padded stride keeps banks spread + 8B align
  const int tid = threadIdx.x;
  const int batch = blockIdx.y;
  const float* Ab = A + (unsigned long long)batch * aStride;
  const float* Bb = B + (unsigned long long)batch * bStride;
  float* Cb = C + (unsigned long long)batch * cStride;

  if (transB) {
    for (int i = tid; i < DIM * DIM; i += 256)        // B[col*128+k] -> Bs[col][k]
      Bs[(i >> 7) * 130 + (i & 127)] = Bb[i];
  } else {
    for (int i = tid; i < DIM * DIM; i += 256)        // B[k*128+col] -> Bs[col][k]
      Bs[(i & 127) * 130 + (i >> 7)] = Bb[i];
  }
  __syncthreads();

  const int lane = tid & 31;
  const int wid = tid >> 5;
  const int tileM = blockIdx.x * 8 + wid;
  if (tileM * 16 < M) {                                // wave-uniform guard
    const int half = lane >> 4;
    const int l15 = lane & 15;
    const int m0 = tileM * 16;
    const unsigned long long arowBase = (unsigned long long)(m0 + l15) * DIM;

    v8f acc[8];
    const v8f zero = {0.f, 0.f, 0.f, 0.f, 0.f, 0.f, 0.f, 0.f};
#pragma unroll
    for (int t = 0; t < 8; ++t) acc[t] = zero;

    for (int k = 0; k < DIM; k += 4) {
      const int kk = k + half * 2;                     // even -> 8B aligned
      v2f a = *(const v2f*)(Ab + arowBase + kk);
#pragma unroll
      for (int t = 0; t < 8; ++t) {
        const int col = t * 16 + l15;
        v2f b = *(const v2f*)(&Bs[col * 130 + kk]);
        acc[t] = __builtin_amdgcn_wmma_f32_16x16x4_f32(
            false, a, false, b, (short)0, acc[t], false, false);
      }
    }
#pragma unroll
    for (int t = 0; t < 8; ++t) {
      const int col = t * 16 + l15;
#pragma unroll
      for (int i = 0; i < 8; ++i)
        Cb[(unsigned long long)(m0 + half * 8 + i) * DIM + col] = acc[t][i];
    }
  }
}

// ---------------------------------------------------------------- GCN aggregate
// out[d] = dinv[d] * sum_{s in N(d)} h[s]*dinv[s]  +  h[d]*dinv[d]^2  +  bias
// one wave per node, float4 per lane (32*4 = 128 dims)
__global__ __launch_bounds__(256) void k_aggregate(
    const float* __restrict__ h, const int* __restrict__ csr,
    const int* __restrict__ offs, const float* __restrict__ dinv,
    const float* __restrict__ bias, float* __restrict__ out) {
  const int node = (blockIdx.x * 256 + threadIdx.x) >> 5;
  const int lane = threadIdx.x & 31;
  if (node >= N_NODES) return;
  const int beg = offs[node], end = offs[node + 1];
  const int c0 = lane * 4;
  float4 acc = make_float4(0.f, 0.f, 0.f, 0.f);
  for (int j = beg; j < end; ++j) {
    const int s = csr[j];
    if (j + 1 < end)
      __builtin_prefetch(h + (unsigned long long)csr[j + 1] * DIM + c0, 0, 0);
    const float w = dinv[s];
    const float4 hv = *(const float4*)(h + (unsigned long long)s * DIM + c0);
    acc.x = fmaf(hv.x, w, acc.x);
    acc.y = fmaf(hv.y, w, acc.y);
    acc.z = fmaf(hv.z, w, acc.z);
    acc.w = fmaf(hv.w, w, acc.w);
  }
  const float di = dinv[node];
  const float4 hs = *(const float4*)(h + (unsigned long long)node * DIM + c0);
  const float4 bv = *(const float4*)(bias + c0);
  float4 o;
  o.x = acc.x * di + hs.x * di * di + bv.x;
  o.y = acc.y * di + hs.y * di * di + bv.y;
  o.z = acc.z * di + hs.z * di * di + bv.z;
  o.w = acc.w * di + hs.w * di * di + bv.w;
  *(float4*)(out + (unsigned long long)node * DIM + c0) = o;
}

// ---------------------------------------------------------------- batch gather
__global__ void k_gather(const float* __restrict__ x, const int* __restrict__ head,
                         const int* __restrict__ tail, float* __restrict__ H,
                         float* __restrict__ T) {
  const int b = blockIdx.x;
  const int t = threadIdx.x;  // 128
  H[b * DIM + t] = x[(unsigned long long)head[b] * DIM + t];
  T[b * DIM + t] = x[(unsigned long long)tail[b] * DIM + t];
}

// ---------------------------------------------------------------- fused score
// out[b, r] = sum_e (H @ Q_r)[b,e] * T[b,e]; wave computes a 16x128 WMMA stripe
// of H@Q_r and immediately reduces it against T (no giant intermediate).
__global__ __launch_bounds__(256) void k_score(
    const float* __restrict__ H, const float* __restrict__ Q,
    const float* __restrict__ T, float* __restrict__ out) {
  __shared__ float Bs[DIM * 130];
  const int tid = threadIdx.x;
  const int r = blockIdx.y;
  const float* Qr = Q + (unsigned long long)r * (DIM * DIM);
  for (int i = tid; i < DIM * DIM; i += 256)
    Bs[(i & 127) * 130 + (i >> 7)] = Qr[i];  // Q[k*128+col] -> Bs[col][k]
  __syncthreads();

  const int lane = tid & 31;
  const int wid = tid >> 5;
  const int m0 = (blockIdx.x * 8 + wid) * 16;  // 2048/16/8 = 16 blocks, always active
  const int half = lane >> 4;
  const int l15 = lane & 15;
  const unsigned long long arowBase = (unsigned long long)(m0 + l15) * DIM;

  v8f acc[8];
  const v8f zero = {0.f, 0.f, 0.f, 0.f, 0.f, 0.f, 0.f, 0.f};
#pragma unroll
  for (int t = 0; t < 8; ++t) acc[t] = zero;

  for (int k = 0; k < DIM; k += 4) {
    const int kk = k + half * 2;
    v2f a = *(const v2f*)(H + arowBase + kk);
#pragma unroll
    for (int t = 0; t < 8; ++t) {
      v2f b = *(const v2f*)(&Bs[(t * 16 + l15) * 130 + kk]);
      acc[t] = __builtin_amdgcn_wmma_f32_16x16x4_f32(
          false, a, false, b, (short)0, acc[t], false, false);
    }
  }

  // row-dot against T: row m0+mi lives in vgpr (mi&7) of lanes with half == mi>>3
#pragma unroll
  for (int mi = 0; mi < 16; ++mi) {
    const int hsel = mi >> 3;
    const int i = mi & 7;
    float p = 0.f;
    if (half == hsel) {
      const float* Trow = T + (unsigned long long)(m0 + mi) * DIM;
#pragma unroll
      for (int t = 0; t < 8; ++t) p = fmaf(acc[t][i], Trow[t * 16 + l15], p);
    }
    p += __shfl_xor(p, 1, 32);
    p += __shfl_xor(p, 2, 32);
    p += __shfl_xor(p, 4, 32);
    p += __shfl_xor(p, 8, 32);
    if (lane == hsel * 16)
      out[(unsigned long long)(m0 + mi) * N_REL + r] = p;
  }
}

// ---------------------------------------------------------------- launch
extern "C" void kernel_launch(void* const* d_in, const int* in_sizes, int n_in,
                              void* d_out, int out_size, void* d_ws, size_t ws_size,
                              hipStream_t stream) {
  const float* init_emb = (const float*)d_in[0];
  const float* W1 = (const float*)d_in[1];
  const float* b1 = (const float*)d_in[2];
  const float* W2 = (const float*)d_in[3];
  const float* b2 = (const float*)d_in[4];
  const float* rel = (const float*)d_in[5];
  const float* inter = (const float*)d_in[6];
  const int* head = (const int*)d_in[7];
  const int* tail = (const int*)d_in[8];
  const int* edge = (const int*)d_in[9];
  const int* src = edge;
  const int* dst = edge + N_EDGES;
  float* out = (float*)d_out;

  char* p = (char*)d_ws;
  auto alloc = [&](size_t bytes) -> void* {
    void* r = (void*)p;
    p += (bytes + 255) & ~(size_t)255;
    return r;
  };
  int* cnt = (int*)alloc((size_t)N_NODES * 4);
  int* offs = (int*)alloc((size_t)(N_NODES + 1) * 4);
  int* cursor = (int*)alloc((size_t)N_NODES * 4);
  int* csr = (int*)alloc((size_t)N_EDGES * 4);
  float* dinv = (float*)alloc((size_t)N_NODES * 4);
  float* h = (float*)alloc((size_t)N_NODES * DIM * 4);
  float* x1 = (float*)alloc((size_t)N_NODES * DIM * 4);
  float* x2 = (float*)alloc((size_t)N_NODES * DIM * 4);
  float* Hemb = (float*)alloc((size_t)BATCH * DIM * 4);
  float* Temb = (float*)alloc((size_t)BATCH * DIM * 4);
  float* P = (float*)alloc((size_t)N_REL * DIM * DIM * 4);
  float* Q = (float*)alloc((size_t)N_REL * DIM * DIM * 4);

  // CSR of incoming edges (built once, used by both layers)
  k_zero_cnt<<<(N_NODES + 255) / 256, 256, 0, stream>>>(cnt);
  k_count<<<(N_EDGES + 255) / 256, 256, 0, stream>>>(dst, cnt);
  k_scan<<<1, 1024, 0, stream>>>(cnt, offs);
  k_dinv_cursor<<<(N_NODES + 255) / 256, 256, 0, stream>>>(cnt, offs, dinv, cursor);
  k_scatter<<<(N_EDGES + 255) / 256, 256, 0, stream>>>(src, dst, cursor, csr);

  // two GCN layers: WMMA GEMM + gather-aggregate
  const int tilesM = (N_NODES + 15) / 16;  // 3125
  dim3 gNodes((tilesM + 7) / 8, 1);
  k_wmma_gemm<<<gNodes, 256, 0, stream>>>(init_emb, 0ULL, W1, 0ULL, 0, h, 0ULL, N_NODES);
  k_aggregate<<<(N_NODES + 7) / 8, 256, 0, stream>>>(h, csr, offs, dinv, b1, x1);
  k_wmma_gemm<<<gNodes, 256, 0, stream>>>(x1, 0ULL, W2, 0ULL, 0, h, 0ULL, N_NODES);
  k_aggregate<<<(N_NODES + 7) / 8, 256, 0, stream>>>(h, csr, offs, dinv, b2, x2);

  // scoring: Q_r = A_r @ M @ A_r^T, then out[b,r] = H[b] @ Q_r . T[b]
  k_gather<<<BATCH, DIM, 0, stream>>>(x2, head, tail, Hemb, Temb);
  k_wmma_gemm<<<dim3(1, N_REL), 256, 0, stream>>>(rel, (unsigned long long)(DIM * DIM),
                                                  inter, 0ULL, 0, P,
                                                  (unsigned long long)(DIM * DIM), DIM);
  k_wmma_gemm<<<dim3(1, N_REL), 256, 0, stream>>>(P, (unsigned long long)(DIM * DIM),
                                                  rel, (unsigned long long)(DIM * DIM), 1, Q,
                                                  (unsigned long long)(DIM * DIM), DIM);
  k_score<<<dim3(BATCH / 128, N_REL), 256, 0, stream>>>(Hemb, Q, Temb, out);
}